// BPICDetector_0_34359738980
// MI455X (gfx1250) — compile-verified
//
#include <hip/hip_runtime.h>
#include <math.h>

// BPIC detector for MI455X (gfx1250, wave32).
// One workgroup (128 thr = 4 waves) per batch item.
//  - H tiles staged into LDS by the Tensor Data Mover (TENSOR_LOAD_TO_LDS),
//    with TDM padding reproducing the [64][33] bank-conflict-free layout.
//  - HtH Gram matrix via V_WMMA_F32_16X16X4_F32 (complex GEMM = 4 real WMMAs
//    per k-step per 16x16 tile); whitening folded into the Gram (1/No).
//  - Gauss-Jordan solve + BPIC iterations in VALU/LDS.

typedef __attribute__((ext_vector_type(2))) float v2f;
typedef __attribute__((ext_vector_type(8))) float v8f;
typedef __attribute__((ext_vector_type(4))) unsigned int v4u;
typedef __attribute__((ext_vector_type(4))) int v4i;
typedef __attribute__((ext_vector_type(8))) int v8i;

#define MM 64
#define KK 32
#define NP 16
#define NITER 5
#define F_EPS 1e-7f
#define F_MINVAR 1e-7f

#if __has_builtin(__builtin_amdgcn_tensor_load_to_lds) && \
    __has_builtin(__builtin_amdgcn_s_wait_tensorcnt)
#define USE_TDM 1
#else
#define USE_TDM 0
#endif

#if USE_TDM
// DMA one 64x32 f32 plane (row-major, stride 32) into LDS laid out as [64][33]
// (TDM pad: +1 DWORD after every 32 DWORDs).  D# per CDNA5 ISA §8.3/8.4.
__device__ __forceinline__ void tdm_load_plane(const float* gsrc, void* lds_dst) {
    unsigned long long ga = (unsigned long long)(uintptr_t)gsrc;
    unsigned lds = (unsigned)(uintptr_t)lds_dst;   // low 32b of flat = LDS offset

    v4u g0;
    g0.x = 1u;                                     // count=1, user desc, no gather
    g0.y = lds;                                    // lds_addr [63:32]
    g0.z = (unsigned)(ga & 0xffffffffu);           // global_addr [95:64]
    g0.w = (unsigned)((ga >> 32) & 0x01ffffffu)    // global_addr [120:96]
         | (2u << 30);                             // type=2 ("image")

    v8i g1;
    g1[0] = (2 << 16)        // data_size = 4 bytes
          | (1 << 20)        // pad_enable
          | (4 << 22);       // pad_interval: 2^(4+1)=32 DWORDs between pads
                             // pad_amount = 0 -> +1 DWORD ; workgroup_mask=0
    g1[1] = (KK << 16);      // tensor_dim0 = 32 (low half in [79:48])
    g1[2] = (MM << 16);      // tensor_dim1 = 64 (low half in [111:80])
    g1[3] = (KK << 16);      // tile_dim0 = 32   ([127:112])
    g1[4] = MM;              // tile_dim1 = 64   ([143:128]); tile_dim2 = 0
    g1[5] = KK;              // tensor_dim0_stride = 32 ([207:160])
    g1[6] = 0;               // stride hi / tensor_dim1_stride lo
    g1[7] = 0;

    v4i z4 = {0, 0, 0, 0};   // groups 2/3 unused for 2-D tensors
#if defined(__clang_major__) && (__clang_major__ >= 23)
    v8i z8 = {0, 0, 0, 0, 0, 0, 0, 0};
    __builtin_amdgcn_tensor_load_to_lds(g0, g1, z4, z4, z8, 0);
#else
    __builtin_amdgcn_tensor_load_to_lds(g0, g1, z4, z4, 0);
#endif
}
#endif

__global__ __launch_bounds__(128) void bpic_kernel(
    const float* __restrict__ gy_re, const float* __restrict__ gy_im,
    const float* __restrict__ gh_re, const float* __restrict__ gh_im,
    const float* __restrict__ gno,
    const float* __restrict__ gp_re, const float* __restrict__ gp_im,
    float* __restrict__ out)
{
    // ---- LDS ----
    __shared__ float Hs_re[MM][33], Hs_im[MM][33];     // RAW H (padded rows)
    __shared__ float ys_re[MM], ys_im[MM];             // raw y
    __shared__ float G_re[KK][33], G_im[KK][33];       // whitened HtH (= raw/No)
    __shared__ float Aug_re[KK][66], Aug_im[KK][66];   // [HtH+I | HtH | Hty]
    __shared__ float fac_re[KK], fac_im[KK];
    __shared__ float Hty_re[KK], Hty_im[KK];
    __shared__ float dg_re[KK], dg_im[KK], dg_rr[KK];  // diag(HtH), 1/Re(diag)
    __shared__ float xd_re[KK], xd_im[KK], vdls[KK];   // DSC state (shared)
    __shared__ float xb_re[KK], xb_im[KK];             // BSE mean (shared)
    __shared__ float p_re[NP], p_im[NP];

    const int b    = blockIdx.x;
    const int t    = threadIdx.x;
    const int lane = t & 31;
    const int wave = t >> 5;

    const float No     = gno[15];          // no[0,0,1,1,1,1]
    const float inv_No = 1.0f / No;        // whitening folded into the Gram

    // ---- stage 1: stage H (TDM), y and constellation into LDS ----
#if USE_TDM
    if (wave == 0) {
        // One wave issues both DMA descriptors (EXEC ignored, TENSORcnt-tracked)
        tdm_load_plane(gh_re + (size_t)b * (MM * KK), &Hs_re[0][0]);
        tdm_load_plane(gh_im + (size_t)b * (MM * KK), &Hs_im[0][0]);
        __builtin_amdgcn_s_wait_tensorcnt(0);
    } else if (wave == 1) {
        ys_re[lane]      = gy_re[(size_t)b * MM + lane];
        ys_re[lane + 32] = gy_re[(size_t)b * MM + lane + 32];
        ys_im[lane]      = gy_im[(size_t)b * MM + lane];
        ys_im[lane + 32] = gy_im[(size_t)b * MM + lane + 32];
    } else if (wave == 2) {
        if (lane < NP) { p_re[lane] = gp_re[lane]; p_im[lane] = gp_im[lane]; }
    }
#else
    {
        const float* hr = gh_re + (size_t)b * (MM * KK);
        const float* hi = gh_im + (size_t)b * (MM * KK);
        for (int i = t; i < MM * KK; i += 128) {
            int m = i >> 5, k = i & 31;
            Hs_re[m][k] = hr[i];
            Hs_im[m][k] = hi[i];
        }
        if (t < MM) {
            ys_re[t] = gy_re[(size_t)b * MM + t];
            ys_im[t] = gy_im[(size_t)b * MM + t];
        }
        if (t < NP) { p_re[t] = gp_re[t]; p_im[t] = gp_im[t]; }
    }
#endif
    __syncthreads();

    // ---- stage 2: HtH = H^H H / No via WMMA f32 16x16x4 ----
    // wave -> 16x16 output tile (i0,j0).  A = H^T tile, B = H tile; both read
    // as plane[m + 2*(lane>>4) + v][col0 + (lane&15)], v = 0,1.
    {
        const int i0 = (wave >> 1) << 4;
        const int j0 = (wave & 1) << 4;
        const int rb = (lane >> 4) << 1;   // k-row base within 4-step
        const int cl = lane & 15;

        v8f accR = {};
        v8f accI = {};
        for (int m = 0; m < MM; m += 4) {
            v2f ar, ai, br, bi, nai;
            ar.x = Hs_re[m + rb][i0 + cl];  ar.y = Hs_re[m + rb + 1][i0 + cl];
            ai.x = Hs_im[m + rb][i0 + cl];  ai.y = Hs_im[m + rb + 1][i0 + cl];
            br.x = Hs_re[m + rb][j0 + cl];  br.y = Hs_re[m + rb + 1][j0 + cl];
            bi.x = Hs_im[m + rb][j0 + cl];  bi.y = Hs_im[m + rb + 1][j0 + cl];
            nai.x = -ai.x; nai.y = -ai.y;   // F32 WMMA has no A-negate (ISA 7.12)

            // re += Hre^T Hre + Him^T Him
            accR = __builtin_amdgcn_wmma_f32_16x16x4_f32(false, ar,  false, br,
                                                         (short)0, accR, false, false);
            accR = __builtin_amdgcn_wmma_f32_16x16x4_f32(false, ai,  false, bi,
                                                         (short)0, accR, false, false);
            // im += Hre^T Him - Him^T Hre
            accI = __builtin_amdgcn_wmma_f32_16x16x4_f32(false, ar,  false, bi,
                                                         (short)0, accI, false, false);
            accI = __builtin_amdgcn_wmma_f32_16x16x4_f32(false, nai, false, br,
                                                         (short)0, accI, false, false);
        }
        const int rh = (lane >> 4) << 3;   // C/D: row = r + 8*(lane>=16)
#pragma unroll
        for (int r = 0; r < 8; ++r) {
            G_re[i0 + r + rh][j0 + cl] = accR[r] * inv_No;
            G_im[i0 + r + rh][j0 + cl] = accI[r] * inv_No;
        }
    }
    __syncthreads();

    // ---- stage 3: Hty (threads 0..31) + augmented-matrix init (threads 32..127) ----
    if (t < KK) {
        float sr = 0.f, si = 0.f;
        for (int m = 0; m < MM; ++m) {
            float hr = Hs_re[m][t], hi = Hs_im[m][t];
            float yr = ys_re[m],    yi = ys_im[m];
            sr += hr * yr + hi * yi;           // conj(h) * y
            si += hr * yi - hi * yr;
        }
        Hty_re[t] = sr * inv_No;  Hty_im[t] = si * inv_No;
        dg_re[t] = G_re[t][t];  dg_im[t] = G_im[t][t];
        dg_rr[t] = 1.0f / G_re[t][t];
    } else {
        int tt = t - 32;
        for (int idx = tt; idx < KK * 64; idx += 96) {
            int r = idx >> 6, c = idx & 63;
            if (c < KK) {
                Aug_re[r][c] = G_re[r][c] + (r == c ? 1.0f : 0.0f);
                Aug_im[r][c] = G_im[r][c];
            } else {
                Aug_re[r][c] = G_re[r][c - KK];
                Aug_im[r][c] = G_im[r][c - KK];
            }
        }
    }
    __syncthreads();
    if (t < KK) { Aug_re[t][64] = Hty_re[t]; Aug_im[t][64] = Hty_im[t]; }
    __syncthreads();

    // ---- stage 4: Gauss-Jordan on [A | HtH | Hty]  (A = HtH+I, Hermitian PD) ----
    for (int p = 0; p < KK; ++p) {
        float pr = Aug_re[p][p], pi = Aug_im[p][p];
        __syncthreads();
        float dn = 1.0f / (pr * pr + pi * pi);
        float ir = pr * dn, ii = -pi * dn;     // 1/pivot
        if (t < 65) {
            float ar = Aug_re[p][t], ai = Aug_im[p][t];
            Aug_re[p][t] = ar * ir - ai * ii;
            Aug_im[p][t] = ar * ii + ai * ir;
        }
        __syncthreads();
        if (t < KK) { fac_re[t] = Aug_re[t][p]; fac_im[t] = Aug_im[t][p]; }
        __syncthreads();
        for (int idx = t; idx < KK * 65; idx += 128) {
            int r = idx / 65, c = idx - r * 65;
            if (r != p) {
                float fr = fac_re[r], fi = fac_im[r];
                float qr = Aug_re[p][c], qi = Aug_im[p][c];
                Aug_re[r][c] -= fr * qr - fi * qi;
                Aug_im[r][c] -= fr * qi + fi * qr;
            }
        }
        __syncthreads();
    }

    // ---- stage 5: BPIC iterations (threads 0..31, one per k) ----
    float x0r = 0.f, x0i = 0.f;
    if (t < KK) {
        // d = diag(inv(A) HtH); z = inv(A) Hty;  x0 = z / d
        float dr = Aug_re[t][KK + t], di = Aug_im[t][KK + t];
        float zr = Aug_re[t][64],     zi = Aug_im[t][64];
        float dn = 1.0f / (dr * dr + di * di);
        x0r = (zr * dr + zi * di) * dn;
        x0i = (zi * dr - zr * di) * dn;
    }

    float ise_prev = 0.f, xbp_r = 0.f, xbp_i = 0.f, vbp = 0.f;
    float xdr = 0.f, xdi = 0.f, vdr = 0.f;
    float xbr = 0.f, xbi = 0.f, vb = 0.f;

    for (int it = 0; it < NITER; ++it) {
        if (t < KK) {
            float xsr, xsi, vbso;
            if (it == 0) {
                xsr = x0r; xsi = x0i;
                vbso = No * dg_rr[t];
            } else {
                float sr = Hty_re[t], si = Hty_im[t], vs = 0.f;
                for (int j = 0; j < KK; ++j) {
                    if (j == t) continue;
                    float gr = G_re[t][j], gi = G_im[t][j];
                    float xr = xd_re[j],   xi = xd_im[j];
                    sr -= gr * xr - gi * xi;
                    si -= gr * xi + gi * xr;
                    vs += (gr * gr + gi * gi) * vdls[j];
                }
                float dr = dg_re[t], di = dg_im[t];
                float dn = 1.0f / (dr * dr + di * di);
                xsr = (sr * dr + si * di) * dn;
                xsi = (si * dr - sr * di) * dn;
                float irr = dg_rr[t];
                vbso = No * irr + vs * irr * irr;
            }
            vbso = fmaxf(vbso, F_MINVAR);

            // BSE: softmax posterior over the 16 constellation points
            float inv2v = 1.0f / (2.0f * vbso);
            float w[NP];
            float mx = -3.0e38f;
#pragma unroll
            for (int q = 0; q < NP; ++q) {
                float er = p_re[q] - xsr, ei = p_im[q] - xsi;
                float l = -(er * er + ei * ei) * inv2v;
                w[q] = l; mx = fmaxf(mx, l);
            }
            float ss = 0.f;
#pragma unroll
            for (int q = 0; q < NP; ++q) { w[q] = expf(w[q] - mx); ss += w[q]; }
            float is = 1.0f / ss;
            xbr = 0.f; xbi = 0.f;
#pragma unroll
            for (int q = 0; q < NP; ++q) {
                float pw = w[q] * is;
                xbr += pw * p_re[q]; xbi += pw * p_im[q]; w[q] = pw;
            }
            vb = 0.f;
#pragma unroll
            for (int q = 0; q < NP; ++q) {
                float er = p_re[q] - xbr, ei = p_im[q] - xbi;
                vb += w[q] * (er * er + ei * ei);
            }
            vb = fmaxf(vb, F_MINVAR);
            xb_re[t] = xbr; xb_im[t] = xbi;
        }
        __syncthreads();

        if (t < KK) {
            // DSC: instantaneous squared error of MRC residual
            float sr = Hty_re[t], si = Hty_im[t];
            for (int j = 0; j < KK; ++j) {
                float gr = G_re[t][j], gi = G_im[t][j];
                float xr = xb_re[j],   xi = xb_im[j];
                sr -= gr * xr - gi * xi;
                si -= gr * xi + gi * xr;
            }
            float dr = dg_re[t], di = dg_im[t];
            float dn = 1.0f / (dr * dr + di * di);
            float er = (sr * dr + si * di) * dn;
            float ei = (si * dr - sr * di) * dn;
            float ise = er * er + ei * ei;

            if (it == 0) { xdr = xbr; xdi = xbi; vdr = vb; }
            else {
                float rho = ise_prev / (ise_prev + ise + F_EPS);
                float om  = 1.0f - rho;
                xdr = om * xbp_r + rho * xbr;
                xdi = om * xbp_i + rho * xbi;
                vdr = om * vbp   + rho * vb;
            }
            ise_prev = ise; xbp_r = xbr; xbp_i = xbi; vbp = vb;
            xd_re[t] = xdr; xd_im[t] = xdi; vdls[t] = vdr;
        }
        __syncthreads();
    }

    // ---- stage 6: maxlog symbol logits ----
    if (t < KK) {
        float iv = 1.0f / fmaxf(vdr, F_MINVAR);
        float* o = out + (size_t)b * (KK * NP) + (size_t)t * NP;
#pragma unroll
        for (int q = 0; q < NP; ++q) {
            float er = p_re[q] - xdr, ei = p_im[q] - xdi;
            o[q] = -(er * er + ei * ei) * iv;
        }
    }
}

extern "C" void kernel_launch(void* const* d_in, const int* in_sizes, int n_in,
                              void* d_out, int out_size, void* d_ws, size_t ws_size,
                              hipStream_t stream) {
    const float* y_re = (const float*)d_in[0];
    const float* y_im = (const float*)d_in[1];
    const float* h_re = (const float*)d_in[2];
    const float* h_im = (const float*)d_in[3];
    const float* no   = (const float*)d_in[4];
    const float* pre  = (const float*)d_in[5];
    const float* pim  = (const float*)d_in[6];
    float* out = (float*)d_out;

    const int B = in_sizes[0] / MM;   // y_re is [B, M]
    bpic_kernel<<<B, 128, 0, stream>>>(y_re, y_im, h_re, h_im, no, pre, pim, out);
}